// CometAttention_34162169872963
// MI455X (gfx1250) — compile-verified
//
#include <hip/hip_runtime.h>
#include <math.h>

typedef __attribute__((ext_vector_type(2))) float v2f;
typedef __attribute__((ext_vector_type(8))) float v8f;

#define B_ 32
#define S_ 2048
#define D_ 1024

// ---------------------------------------------------------------------------
// Phase 1: scores[b][s] = dot(query[b], values[b][s]) * (1/sqrt(D))
// One wave computes a 16-row s-tile via V_WMMA_F32_16X16X4_F32:
//   A (16x4)  = query fragment broadcast down M
//   B (4x16)  = values tile, column n = row s0+n
//   D column n -> score(s0+n), identical across M
// Block = 256 threads (8 waves), each block: one b, 128 s rows.
// ---------------------------------------------------------------------------
__global__ __launch_bounds__(256) void ca_scores_kernel(
    const float* __restrict__ q, const float* __restrict__ values,
    float* __restrict__ scores) {
  __shared__ float qsh[D_];
  __shared__ float tile[8][16][68];  // 68-float row pitch: conflict-free reads

  const int tid = threadIdx.x;
  const int lane = tid & 31;
  const int wv = tid >> 5;
  const int b = blockIdx.x >> 4;            // 32 b's
  const int sBase = (blockIdx.x & 15) << 7; // 16 chunks of 128 s
  const int s0 = sBase + wv * 16;

  // Stage query row for this b (all 8 waves share b): 256 thr * float4 = 1024f
  ((float4*)qsh)[tid] = ((const float4*)(q + (size_t)b * D_))[tid];
  __syncthreads();

  const int kh2 = (lane >> 4) << 1;  // 0 for lanes 0-15, 2 for lanes 16-31
  v8f c = {};

  const float* vbase = values + ((size_t)b * S_ + s0) * D_;

  for (int d0 = 0; d0 < D_; d0 += 64) {
    // Coalesced stage of this wave's 16x64 fp32 values tile into LDS.
    // Same-wave LDS ordering (DScnt in-order) makes this barrier-free.
#pragma unroll
    for (int t = 0; t < 8; ++t) {
      const int idx = t * 32 + lane;
      const int row = idx >> 4;  // 0..15
      const int c4 = idx & 15;   // 0..15 (float4 column)
      const float4 vv =
          *(const float4*)(vbase + (size_t)row * D_ + d0 + (c4 << 2));
      *(float4*)&tile[wv][row][c4 << 2] = vv;
    }
#pragma unroll
    for (int dd = 0; dd < 64; dd += 4) {
      // A fragment: q pair (K = kh2, kh2+1) — broadcast across M
      const float2 av = *(const float2*)&qsh[d0 + dd + kh2];
      // B fragment: values pair for column N = lane&15
      const float2 bv = *(const float2*)&tile[wv][lane & 15][dd + kh2];
      v2f a;  a.x = av.x;  a.y = av.y;
      v2f bb; bb.x = bv.x; bb.y = bv.y;
      c = __builtin_amdgcn_wmma_f32_16x16x4_f32(
          false, a, false, bb, (short)0, c, false, false);
    }
  }

  // D layout: VGPR0, lanes 0-15 -> (M=0, N=lane); column N holds score(s0+N).
  if (lane < 16) {
    scores[(size_t)b * S_ + s0 + lane] = c[0] * 0.03125f;  // 1/sqrt(1024)
  }
}

// ---------------------------------------------------------------------------
// Phase 2: softmax over the batch axis (32 entries per s == one wave32).
// lane = b; cross-lane max/sum via shfl_xor. In-place: each wave exclusively
// owns the 32 entries of its s.
// ---------------------------------------------------------------------------
__global__ __launch_bounds__(256) void ca_softmax_b_kernel(float* scores) {
  const int t = blockIdx.x * 256 + threadIdx.x;  // 65536 = 32*2048
  const int s = t >> 5;
  const int b = t & 31;
  const float x = scores[(size_t)b * S_ + s];

  float m = x;
#pragma unroll
  for (int off = 16; off >= 1; off >>= 1)
    m = fmaxf(m, __shfl_xor(m, off, 32));
  const float e = expf(x - m);
  float sum = e;
#pragma unroll
  for (int off = 16; off >= 1; off >>= 1)
    sum += __shfl_xor(sum, off, 32);

  scores[(size_t)b * S_ + s] = e / sum;
}

// ---------------------------------------------------------------------------
// Phase 3: out[b,s,:] = values[b,s,:] * w[b,s]. Pure streaming, float4 wide.
// Row (b*S+s) = float4_index >> 8 since D/4 = 256.
// ---------------------------------------------------------------------------
__global__ __launch_bounds__(256) void ca_scale_kernel(
    const float* __restrict__ values, const float* __restrict__ weights,
    float* __restrict__ out) {
  const size_t i = (size_t)blockIdx.x * 256 + threadIdx.x;  // float4 index
  const float4 v = ((const float4*)values)[i];
  const float w = weights[i >> 8];
  float4 o;
  o.x = v.x * w; o.y = v.y * w; o.z = v.z * w; o.w = v.w * w;
  ((float4*)out)[i] = o;
}

extern "C" void kernel_launch(void* const* d_in, const int* in_sizes, int n_in,
                              void* d_out, int out_size, void* d_ws,
                              size_t ws_size, hipStream_t stream) {
  (void)in_sizes; (void)n_in; (void)out_size; (void)ws_size;
  const float* q = (const float*)d_in[0];       // [B, D]
  const float* values = (const float*)d_in[1];  // [B, S, D]
  float* out = (float*)d_out;                   // [B, S, D]
  float* scores = (float*)d_ws;                 // [B, S] fp32 = 256 KB

  // 1) scores: 512 blocks = 32 b * 16 s-chunks, 256 threads (8 waves)
  ca_scores_kernel<<<512, 256, 0, stream>>>(q, values, scores);
  // 2) softmax over b: 65536 threads, one wave per s
  ca_softmax_b_kernel<<<256, 256, 0, stream>>>(scores);
  // 3) scale: 16M float4 elements
  ca_scale_kernel<<<65536, 256, 0, stream>>>(values, scores, out);
}